// PairwiseInteraction_50337016709353
// MI455X (gfx1250) — compile-verified
//
#include <hip/hip_runtime.h>

typedef __attribute__((ext_vector_type(2))) float v2f;
typedef __attribute__((ext_vector_type(8))) float v8f;

constexpr int Bsz = 128;
constexpr int Nsl = 32;
constexpr int Dd  = 128;
constexpr int Hh  = 256;
constexpr int BN  = Bsz * Nsl;          // 4096 rows
constexpr float LN_EPS = 1e-5f;

// ---------------------------------------------------------------------------
// Kernel 1: [U|V] = slots(4096x128) @ [W_enc_top | W_enc_bot]  -> 4096x512
// One wave computes one 16x16 f32 tile via 32x V_WMMA_F32_16X16X4_F32 (K=128).
// ---------------------------------------------------------------------------
__global__ void enc_gemm_kernel(const float* __restrict__ slots,
                                const float* __restrict__ Wenc,   // (256,256) row-major
                                float* __restrict__ UV)           // U: [0,4096*256), V: next
{
    const int lane = threadIdx.x & 31;
    const int wave = threadIdx.x >> 5;
    const int tile = blockIdx.x * 8 + wave;    // 0..8191
    const int tm = tile >> 5;                  // 256 M-tiles
    const int tn = tile & 31;                  // 32 combined-N tiles (512 cols)

    const int row  = tm * 16 + (lane & 15);    // A row (lanes 0-15 & 16-31 same M)
    const int cg   = tn * 16 + (lane & 15);    // combined col 0..511
    const int whichW = cg >> 8;                // 0 -> U (W rows 0..127), 1 -> V (128..255)
    const int hcol   = cg & 255;
    const int klo  = (lane >> 4) * 2;          // K sub-offset per ISA f32 A/B layout

    const float* __restrict__ arow = slots + (size_t)row * Dd;

    v8f acc = {};
    #pragma unroll 4
    for (int k = 0; k < Dd; k += 4) {
        const int kk = k + klo;
        v2f a; a.x = arow[kk]; a.y = arow[kk + 1];
        const float* bp = Wenc + (size_t)(whichW * Dd + kk) * Hh + hcol;
        v2f b; b.x = bp[0]; b.y = bp[Hh];
        acc = __builtin_amdgcn_wmma_f32_16x16x4_f32(false, a, false, b,
                                                    (short)0, acc, false, false);
    }

    const int hi = lane >> 4;
    float* __restrict__ dst = UV + (size_t)whichW * BN * Hh;
    #pragma unroll
    for (int r = 0; r < 8; ++r) {
        const int rowm = tm * 16 + r + 8 * hi;
        dst[(size_t)rowm * Hh + hcol] = acc[r];
    }
}

// ---------------------------------------------------------------------------
// Kernel 2: per (b,i): for j != i
//   h = relu( LN(U[b,i] + V[b,j] + b_enc) * ln_g + ln_b )
//   a = sigmoid( h . W_att + b_att )
//   G[b,i] += a*h ;  S[b,i] += a
// One block (256 thr = 8 waves) per (b,i); each wave owns j = wave + 8t.
// Lane l holds h-elements {l + 32*t}, t<8 (coalesced). Wave-only reductions.
// ---------------------------------------------------------------------------
__device__ inline float wave_reduce_sum(float v) {
    #pragma unroll
    for (int m = 16; m >= 1; m >>= 1) v += __shfl_xor(v, m, 32);
    return v;  // full butterfly: all lanes hold the sum
}

__global__ void pair_reduce_kernel(const float* __restrict__ U,
                                   const float* __restrict__ V,
                                   const float* __restrict__ b_enc,
                                   const float* __restrict__ ln_g,
                                   const float* __restrict__ ln_b,
                                   const float* __restrict__ W_att,
                                   const float* __restrict__ b_att,
                                   float* __restrict__ G,
                                   float* __restrict__ S)
{
    __shared__ float gsh[8 * Hh];
    __shared__ float ssh[8];

    const int bi   = blockIdx.x;           // 0..4095
    const int b    = bi >> 5;
    const int i    = bi & 31;
    const int lane = threadIdx.x & 31;
    const int wave = threadIdx.x >> 5;

    const float batt = b_att[0];

    float u[8], be[8], lg[8], lb[8], wa[8], gacc[8];
    #pragma unroll
    for (int t = 0; t < 8; ++t) {
        const int idx = lane + 32 * t;
        u[t]  = U[(size_t)bi * Hh + idx];
        be[t] = b_enc[idx];
        lg[t] = ln_g[idx];
        lb[t] = ln_b[idx];
        wa[t] = W_att[idx];
        gacc[t] = 0.0f;
    }
    float sacc = 0.0f;

    for (int jj = wave; jj < Nsl; jj += 8) {
        if (jj == i) continue;                         // wave-uniform skip
        const float* __restrict__ vrow = V + (size_t)(b * Nsl + jj) * Hh;

        float h[8];
        float sum = 0.0f;
        #pragma unroll
        for (int t = 0; t < 8; ++t) {
            h[t] = u[t] + vrow[lane + 32 * t] + be[t];
            sum += h[t];
        }
        const float mu = wave_reduce_sum(sum) * (1.0f / Hh);

        float sq = 0.0f;
        #pragma unroll
        for (int t = 0; t < 8; ++t) { const float d = h[t] - mu; sq += d * d; }
        const float var = wave_reduce_sum(sq) * (1.0f / Hh);
        const float inv = rsqrtf(var + LN_EPS);

        float adot = 0.0f;
        #pragma unroll
        for (int t = 0; t < 8; ++t) {
            float hn = (h[t] - mu) * inv * lg[t] + lb[t];
            hn = fmaxf(hn, 0.0f);                      // ReLU
            h[t] = hn;
            adot += hn * wa[t];
        }
        adot = wave_reduce_sum(adot) + batt;
        const float a = 1.0f / (1.0f + __expf(-adot)); // sigmoid

        #pragma unroll
        for (int t = 0; t < 8; ++t) gacc[t] += a * h[t];
        sacc += a;
    }

    #pragma unroll
    for (int t = 0; t < 8; ++t) gsh[wave * Hh + lane + 32 * t] = gacc[t];
    if (lane == 0) ssh[wave] = sacc;
    __syncthreads();

    float tot = 0.0f;
    #pragma unroll
    for (int w = 0; w < 8; ++w) tot += gsh[w * Hh + threadIdx.x];
    G[(size_t)bi * Hh + threadIdx.x] = tot;

    if (threadIdx.x == 0) {
        float st = 0.0f;
        #pragma unroll
        for (int w = 0; w < 8; ++w) st += ssh[w];
        S[bi] = st;
    }
}

// ---------------------------------------------------------------------------
// Kernel 3: out(4096x128) = G(4096x256) @ W_eff(256x128) + S[row]*b_eff[col]
// One wave per 16x16 tile, 64x V_WMMA_F32_16X16X4_F32 (K=256).
// ---------------------------------------------------------------------------
__global__ void eff_gemm_kernel(const float* __restrict__ G,
                                const float* __restrict__ S,
                                const float* __restrict__ Weff,   // (256,128)
                                const float* __restrict__ beff,   // (128,)
                                float* __restrict__ out)          // (4096,128)
{
    const int lane = threadIdx.x & 31;
    const int wave = threadIdx.x >> 5;
    const int tile = blockIdx.x * 8 + wave;   // 0..2047
    const int tm = tile >> 3;                 // 256 M-tiles
    const int tn = tile & 7;                  // 8 N-tiles (128 cols)

    const int row = tm * 16 + (lane & 15);
    const int col = tn * 16 + (lane & 15);
    const int klo = (lane >> 4) * 2;

    const float* __restrict__ arow = G + (size_t)row * Hh;

    v8f acc = {};
    #pragma unroll 4
    for (int k = 0; k < Hh; k += 4) {
        const int kk = k + klo;
        v2f a; a.x = arow[kk]; a.y = arow[kk + 1];
        const float* bp = Weff + (size_t)kk * Dd + col;
        v2f b; b.x = bp[0]; b.y = bp[Dd];
        acc = __builtin_amdgcn_wmma_f32_16x16x4_f32(false, a, false, b,
                                                    (short)0, acc, false, false);
    }

    const float bec = beff[col];
    const int hi = lane >> 4;
    #pragma unroll
    for (int r = 0; r < 8; ++r) {
        const int rowm = tm * 16 + r + 8 * hi;
        out[(size_t)rowm * Dd + col] = acc[r] + S[rowm] * bec;
    }
}

// ---------------------------------------------------------------------------
extern "C" void kernel_launch(void* const* d_in, const int* in_sizes, int n_in,
                              void* d_out, int out_size, void* d_ws, size_t ws_size,
                              hipStream_t stream)
{
    const float* slots = (const float*)d_in[0];   // (128,32,128)
    const float* W_enc = (const float*)d_in[1];   // (256,256)
    const float* b_enc = (const float*)d_in[2];   // (256,)
    const float* ln_g  = (const float*)d_in[3];   // (256,)
    const float* ln_b  = (const float*)d_in[4];   // (256,)
    const float* W_att = (const float*)d_in[5];   // (256,1)
    const float* b_att = (const float*)d_in[6];   // (1,)
    const float* W_eff = (const float*)d_in[7];   // (256,128)
    const float* b_eff = (const float*)d_in[8];   // (128,)
    float* out = (float*)d_out;                   // (128,32,128)

    float* ws = (float*)d_ws;
    float* U  = ws;                                // 4096*256
    float* V  = U + (size_t)BN * Hh;               // 4096*256
    float* Gm = V + (size_t)BN * Hh;               // 4096*256
    float* Sm = Gm + (size_t)BN * Hh;              // 4096

    // 1) U,V = slots @ [W_top | W_bot]: 8192 tiles, 8 waves/block
    enc_gemm_kernel<<<1024, 256, 0, stream>>>(slots, W_enc, U);

    // 2) pairwise LN/attn reduction -> G, S: one block per (b,i)
    pair_reduce_kernel<<<BN, 256, 0, stream>>>(U, V, b_enc, ln_g, ln_b,
                                               W_att, b_att, Gm, Sm);

    // 3) out = G @ W_eff + S*b_eff: 2048 tiles, 8 waves/block
    eff_gemm_kernel<<<256, 256, 0, stream>>>(Gm, Sm, W_eff, b_eff, out);
}